// DocumentGAT_26285199851933
// MI455X (gfx1250) — compile-verified
//
#include <hip/hip_runtime.h>
#include <hip/hip_bf16.h>

// ---------------------------------------------------------------------------
// DocumentGAT for MI455X (gfx1250, wave32).
//  - GEMMs via V_WMMA_F32_16X16X4_F32 (fp32; both GEMMs are bandwidth-bound).
//  - B staged TRANSPOSED in LDS so each WMMA B operand is one ds_load_b64
//    (no VGPR repacking), mirroring the A path.
//  - Edge softmax via order-preserving uint atomicMax + f32 atomicAdd.
// ---------------------------------------------------------------------------

typedef __attribute__((ext_vector_type(2))) float v2f;
typedef __attribute__((ext_vector_type(8))) float v8f;

#define NEG_SLOPE 0.2f
#define ENC_NEG_INF 0x007FFFFFu   // fenc(-inf)

__device__ __forceinline__ unsigned fenc(float f) {
  unsigned u = __float_as_uint(f);
  return (u & 0x80000000u) ? ~u : (u | 0x80000000u);
}
__device__ __forceinline__ float fdec(unsigned u) {
  unsigned v = (u & 0x80000000u) ? (u & 0x7FFFFFFFu) : ~u;
  return __uint_as_float(v);
}
__device__ __forceinline__ float elu_f(float v) {
  return v > 0.f ? v : (__expf(v) - 1.f);
}
__device__ __forceinline__ void edge_sd(int e, const int* __restrict__ ei, int E,
                                        int& s, int& d) {
  if (e < E) { s = ei[e]; d = ei[E + e]; }
  else       { s = e - E; d = e - E; }      // appended self-loops
}

// ---------------------------------------------------------------------------
// fill kernel (replaces hipMemsetAsync; graph-capture safe, deterministic)
// ---------------------------------------------------------------------------
__global__ void fill_u32(unsigned* __restrict__ p, unsigned v, size_t n) {
  size_t i = (size_t)blockIdx.x * blockDim.x + threadIdx.x;
  if (i < n) p[i] = v;
}

// ---------------------------------------------------------------------------
// C[M,64] = A[M,K] @ B[K,64]   (fp32, WMMA 16x16x4)
// 256 threads = 8 waves; block tile 128x64; K staged 32 wide through LDS.
// Wave w owns rows [w*16, w*16+16) of the tile and all four 16-col subtiles.
//   A fragment (16x4 f32): lanes 0-15 hold K,K+1; lanes 16-31 hold K+2,K+3.
//   B fragment mirrored; Bs is stored transposed so both fragment dwords are
//   adjacent in LDS -> single ds_load_b64 per operand.
// ---------------------------------------------------------------------------
__global__ __launch_bounds__(256) void gemm64_wmma(
    const float* __restrict__ A, const float* __restrict__ B,
    float* __restrict__ C, int M, int K) {
  constexpr int MT = 128, KC = 32, NW = 64;
  constexpr int ASTR = KC + 4;   // 36: conflict-free column reads (16 rows)
  constexpr int BSTR = KC + 2;   // 34: conflict-free col reads, 8B-aligned k-pairs
  __shared__ float As[MT][ASTR];
  __shared__ float Bs[NW][BSTR];            // transposed: [col][k]

  const int tid  = threadIdx.x;
  const int wid  = tid >> 5;
  const int lane = tid & 31;
  const int l16  = lane & 15;
  const int hi   = (lane >> 4) & 1;         // upper half-wave -> K+2,K+3
  const int rowBase = blockIdx.x * MT;

  v8f acc[4] = {};                          // four 16x16 f32 accumulators

  for (int k0 = 0; k0 < K; k0 += KC) {
    // ---- stage A tile (128 x 32), float4 per unit, zero-fill OOB ----
    for (int u = tid; u < (MT * KC) / 4; u += 256) {
      int r  = u >> 3;
      int c4 = (u & 7) << 2;
      float4 v = {0.f, 0.f, 0.f, 0.f};
      int gr = rowBase + r;
      if (gr < M && (k0 + c4) < K)          // K is a multiple of 4
        v = *reinterpret_cast<const float4*>(A + (size_t)gr * K + k0 + c4);
      As[r][c4 + 0] = v.x; As[r][c4 + 1] = v.y;
      As[r][c4 + 2] = v.z; As[r][c4 + 3] = v.w;
    }
    // ---- stage B tile (32 x 64) transposed into Bs[col][k] ----
    for (int u = tid; u < (KC * NW) / 4; u += 256) {
      int r  = u >> 4;                      // k within chunk
      int c4 = (u & 15) << 2;               // starting column
      float4 v = {0.f, 0.f, 0.f, 0.f};
      if (k0 + r < K)
        v = *reinterpret_cast<const float4*>(B + (size_t)(k0 + r) * NW + c4);
      Bs[c4 + 0][r] = v.x; Bs[c4 + 1][r] = v.y;
      Bs[c4 + 2][r] = v.z; Bs[c4 + 3][r] = v.w;
    }
    // prefetch next A chunk (global_prefetch_b8; speculative, drop on fault)
    if (k0 + KC < K) {
      int pr = rowBase + (tid >> 1);
      if (pr < M) __builtin_prefetch(A + (size_t)pr * K + k0 + KC, 0, 0);
    }
    __syncthreads();

    const int myRow = (wid << 4) + l16;
#pragma unroll
    for (int kk = 0; kk < KC; kk += 4) {
      const int kp = kk + (hi << 1);        // lanes 0-15: kk; 16-31: kk+2
      v2f a = *reinterpret_cast<const v2f*>(&As[myRow][kp]);   // ds_load_b64
#pragma unroll
      for (int n = 0; n < 4; ++n) {
        v2f b = *reinterpret_cast<const v2f*>(&Bs[(n << 4) + l16][kp]); // b64
        acc[n] = __builtin_amdgcn_wmma_f32_16x16x4_f32(
            false, a, false, b, (short)0, acc[n], false, false);
      }
    }
    __syncthreads();
  }

  // ---- store: VGPR r holds row M=r (lanes 0-15) / M=r+8 (lanes 16-31) ----
#pragma unroll
  for (int n = 0; n < 4; ++n) {
#pragma unroll
    for (int r = 0; r < 8; ++r) {
      int row = rowBase + (wid << 4) + (hi ? r + 8 : r);
      if (row < M) C[(size_t)row * NW + (n << 4) + l16] = acc[n][r];
    }
  }
}

// ---------------------------------------------------------------------------
// per-node attention logits:  al_src[n,h] = <h[n,h,:], a_src[h,:]>  (ditto dst)
// ---------------------------------------------------------------------------
__global__ void node_alpha(const float* __restrict__ h,
                           const float* __restrict__ a_src,
                           const float* __restrict__ a_dst,
                           float* __restrict__ al_s, float* __restrict__ al_d,
                           int Nn, int H, int C) {
  int i = blockIdx.x * blockDim.x + threadIdx.x;
  if (i >= Nn * H) return;
  int n = i / H, hh = i - n * H;
  const float* row = h + (size_t)n * H * C + (size_t)hh * C;
  float s = 0.f, d = 0.f;
  for (int c = 0; c < C; ++c) {
    float v = row[c];
    s += v * a_src[hh * C + c];
    d += v * a_dst[hh * C + c];
  }
  al_s[i] = s; al_d[i] = d;
}

// ---------------------------------------------------------------------------
// edge pass 1: segment max of leaky_relu(al_src[src]+al_dst[dst]) over dst
// ---------------------------------------------------------------------------
__global__ void edge_max(const int* __restrict__ ei, int E, int ET,
                         const float* __restrict__ als,
                         const float* __restrict__ ald,
                         unsigned* __restrict__ emaxu, int H) {
  int i = blockIdx.x * blockDim.x + threadIdx.x;
  if (i >= ET * H) return;
  int e = i / H, hh = i - e * H;
  int s, d; edge_sd(e, ei, E, s, d);
  float v = als[s * H + hh] + ald[d * H + hh];
  v = v > 0.f ? v : NEG_SLOPE * v;
  atomicMax(&emaxu[d * H + hh], fenc(v));
}

// ---------------------------------------------------------------------------
// edge pass 2: ex = exp(e - max[dst]); denom[dst] += ex; stash ex per edge
// ---------------------------------------------------------------------------
__global__ void edge_exp(const int* __restrict__ ei, int E, int ET,
                         const float* __restrict__ als,
                         const float* __restrict__ ald,
                         const unsigned* __restrict__ emaxu,
                         float* __restrict__ ex, float* __restrict__ denom,
                         int H) {
  int i = blockIdx.x * blockDim.x + threadIdx.x;
  if (i >= ET * H) return;
  int e = i / H, hh = i - e * H;
  int s, d; edge_sd(e, ei, E, s, d);
  float v = als[s * H + hh] + ald[d * H + hh];
  v = v > 0.f ? v : NEG_SLOPE * v;
  float m = fdec(emaxu[d * H + hh]);
  float x = __expf(v - m);
  ex[i] = x;
  atomicAdd(&denom[d * H + hh], x);
}

// ---------------------------------------------------------------------------
// edge pass 3: out[dst,c] += alpha * h[src,c]   (one thread per edge-channel)
// ---------------------------------------------------------------------------
__global__ void edge_agg(const int* __restrict__ ei, int E, int ET,
                         const float* __restrict__ ex,
                         const float* __restrict__ denom,
                         const float* __restrict__ hsrc,
                         float* __restrict__ outp, int H, int C) {
  long long i = (long long)blockIdx.x * blockDim.x + threadIdx.x;
  const int HC = H * C;
  if (i >= (long long)ET * HC) return;
  int e = (int)(i / HC);
  int c = (int)(i - (long long)e * HC);
  int hh = c / C;
  int s, d; edge_sd(e, ei, E, s, d);
  float alpha = ex[(size_t)e * H + hh] / (denom[d * H + hh] + 1e-16f);
  atomicAdd(&outp[(size_t)d * HC + c], alpha * hsrc[(size_t)s * HC + c]);
}

// ---------------------------------------------------------------------------
// h = elu(in + bias)
// ---------------------------------------------------------------------------
__global__ void bias_elu(const float* __restrict__ in, const float* __restrict__ b,
                         float* __restrict__ outp, int Nn, int F) {
  int i = blockIdx.x * blockDim.x + threadIdx.x;
  if (i >= Nn * F) return;
  int f = i % F;
  outp[i] = elu_f(in[i] + b[f]);
}

// ---------------------------------------------------------------------------
// layer-2 epilogue fused with global mean pool accumulation
// ---------------------------------------------------------------------------
__global__ void bias_elu_pool(const float* __restrict__ in,
                              const float* __restrict__ b,
                              const int* __restrict__ batch,
                              float* __restrict__ pooled, float* __restrict__ cnt,
                              int Nn, int F) {
  int i = blockIdx.x * blockDim.x + threadIdx.x;
  if (i >= Nn * F) return;
  int n = i / F, f = i - n * F;
  float v = elu_f(in[i] + b[f]);
  int g = batch[n];
  atomicAdd(&pooled[(size_t)g * F + f], v);
  if (f == 0) atomicAdd(&cnt[g], 1.0f);
}

// ---------------------------------------------------------------------------
// pooled mean -> logits -> log_softmax     (one thread per graph; tiny)
// ---------------------------------------------------------------------------
__global__ void classify(const float* __restrict__ pooled,
                         const float* __restrict__ cnt,
                         const float* __restrict__ Wc,
                         const float* __restrict__ bc,
                         float* __restrict__ outp, int G, int F, int NC) {
  int g = blockIdx.x * blockDim.x + threadIdx.x;
  if (g >= G) return;
  float inv = 1.0f / fmaxf(cnt[g], 1.0f);
  float logits[32];
  for (int j = 0; j < NC; ++j) logits[j] = bc[j];
  for (int i = 0; i < F; ++i) {
    float p = pooled[(size_t)g * F + i] * inv;
    for (int j = 0; j < NC; ++j) logits[j] += p * Wc[i * NC + j];
  }
  float m = -INFINITY;
  for (int j = 0; j < NC; ++j) m = fmaxf(m, logits[j]);
  float ssum = 0.f;
  for (int j = 0; j < NC; ++j) ssum += __expf(logits[j] - m);
  float lse = logf(ssum);
  for (int j = 0; j < NC; ++j) outp[(size_t)g * NC + j] = logits[j] - m - lse;
}

// ---------------------------------------------------------------------------
extern "C" void kernel_launch(void* const* d_in, const int* in_sizes, int n_in,
                              void* d_out, int out_size, void* d_ws, size_t ws_size,
                              hipStream_t stream) {
  const float* x      = (const float*)d_in[0];
  const int*   ei     = (const int*)d_in[1];
  const int*   batch  = (const int*)d_in[2];
  const float* W1     = (const float*)d_in[3];
  const float* a_src1 = (const float*)d_in[4];
  const float* a_dst1 = (const float*)d_in[5];
  const float* b1     = (const float*)d_in[6];
  const float* W2     = (const float*)d_in[7];
  const float* a_src2 = (const float*)d_in[8];
  const float* a_dst2 = (const float*)d_in[9];
  const float* b2     = (const float*)d_in[10];
  const float* Wc     = (const float*)d_in[11];
  const float* bc     = (const float*)d_in[12];
  float* out = (float*)d_out;

  const int N   = in_sizes[2];          // 50000 nodes
  const int E   = in_sizes[1] / 2;      // 1.6M edges
  const int ET  = E + N;                // + self-loops
  const int HID = in_sizes[6];          // 64
  const int VOC = in_sizes[0] / N;      // 5000
  const int H1  = 4;
  const int C1  = HID / H1;             // 16
  const int NC  = in_sizes[12];         // 20
  const int G   = 128;                  // NUM_GRAPHS

  // ---- carve workspace (aliased to stay ~66 MB) ----
  size_t off = 0;
  auto carve = [&](size_t bytes) -> void* {
    void* p = (char*)d_ws + off;
    off = (off + bytes + 255) & ~(size_t)255;
    return p;
  };
  float*    h1    = (float*)carve((size_t)N * HID * 4);  // layer1 h; reused as layer2 h
  float*    hl1   = (float*)carve((size_t)N * HID * 4);  // elu(layer1)
  float*    outA  = (float*)carve((size_t)N * HID * 4);  // agg buffer (layer1 then layer2)
  float*    exb   = (float*)carve((size_t)ET * H1 * 4);  // edge exp (layer1; reused layer2)
  float*    als1  = (float*)carve((size_t)N * H1 * 4);
  float*    ald1  = (float*)carve((size_t)N * H1 * 4);
  unsigned* emax1 = (unsigned*)carve((size_t)N * H1 * 4);
  float*    den1  = (float*)carve((size_t)N * H1 * 4);
  float*    als2  = (float*)carve((size_t)N * 4);
  float*    ald2  = (float*)carve((size_t)N * 4);
  unsigned* emax2 = (unsigned*)carve((size_t)N * 4);
  float*    den2  = (float*)carve((size_t)N * 4);
  float*    pooled= (float*)carve((size_t)G * HID * 4);
  float*    cnt   = (float*)carve((size_t)G * 4);
  (void)ws_size; (void)n_in; (void)out_size;

  auto cdiv = [](long long a, long long b) -> unsigned { return (unsigned)((a + b - 1) / b); };

  // ---- init accumulators ----
  fill_u32<<<cdiv((long long)N * HID, 256), 256, 0, stream>>>((unsigned*)outA, 0u, (size_t)N * HID);
  fill_u32<<<cdiv((long long)N * H1, 256), 256, 0, stream>>>(emax1, ENC_NEG_INF, (size_t)N * H1);
  fill_u32<<<cdiv((long long)N * H1, 256), 256, 0, stream>>>((unsigned*)den1, 0u, (size_t)N * H1);
  fill_u32<<<cdiv(N, 256), 256, 0, stream>>>(emax2, ENC_NEG_INF, (size_t)N);
  fill_u32<<<cdiv(N, 256), 256, 0, stream>>>((unsigned*)den2, 0u, (size_t)N);
  fill_u32<<<cdiv((long long)G * HID, 256), 256, 0, stream>>>((unsigned*)pooled, 0u, (size_t)G * HID);
  fill_u32<<<cdiv(G, 256), 256, 0, stream>>>((unsigned*)cnt, 0u, (size_t)G);

  // ---- layer 1 ----
  gemm64_wmma<<<cdiv(N, 128), 256, 0, stream>>>(x, W1, h1, N, VOC);
  node_alpha<<<cdiv((long long)N * H1, 256), 256, 0, stream>>>(h1, a_src1, a_dst1, als1, ald1, N, H1, C1);
  edge_max<<<cdiv((long long)ET * H1, 256), 256, 0, stream>>>(ei, E, ET, als1, ald1, emax1, H1);
  edge_exp<<<cdiv((long long)ET * H1, 256), 256, 0, stream>>>(ei, E, ET, als1, ald1, emax1, exb, den1, H1);
  edge_agg<<<cdiv((long long)ET * HID, 256), 256, 0, stream>>>(ei, E, ET, exb, den1, h1, outA, H1, C1);
  bias_elu<<<cdiv((long long)N * HID, 256), 256, 0, stream>>>(outA, b1, hl1, N, HID);

  // re-zero agg buffer for layer 2 (outA reused)
  fill_u32<<<cdiv((long long)N * HID, 256), 256, 0, stream>>>((unsigned*)outA, 0u, (size_t)N * HID);

  // ---- layer 2 (single head, C = HID); h2 aliases h1 ----
  float* h2 = h1;
  float* ex2 = exb;
  gemm64_wmma<<<cdiv(N, 128), 256, 0, stream>>>(hl1, W2, h2, N, HID);
  node_alpha<<<cdiv(N, 256), 256, 0, stream>>>(h2, a_src2, a_dst2, als2, ald2, N, 1, HID);
  edge_max<<<cdiv(ET, 256), 256, 0, stream>>>(ei, E, ET, als2, ald2, emax2, 1);
  edge_exp<<<cdiv(ET, 256), 256, 0, stream>>>(ei, E, ET, als2, ald2, emax2, ex2, den2, 1);
  edge_agg<<<cdiv((long long)ET * HID, 256), 256, 0, stream>>>(ei, E, ET, ex2, den2, h2, outA, 1, HID);
  bias_elu_pool<<<cdiv((long long)N * HID, 256), 256, 0, stream>>>(outA, b2, batch, pooled, cnt, N, HID);

  // ---- classifier + log_softmax ----
  classify<<<cdiv(G, 64), 64, 0, stream>>>(pooled, cnt, Wc, bc, out, G, HID, NC);
}